// NeuMipMap_42975442764261
// MI455X (gfx1250) — compile-verified
//
#include <hip/hip_runtime.h>
#include <hip/hip_bf16.h>
#include <stdint.h>

#define CH     8
#define NLVL   5
#define HW     (1024 * 1024)
#define NPIX   (4 * HW)

// Channel-last workspace offsets (floats): off(l) = sum_{k<l} 8*(512>>k)^2
//   = (8388608 - (8388608 >> (2l))) / 3
__device__ __forceinline__ int ws_off(int l) {
    return (8388608 - (8388608 >> (2 * l))) / 3;
}
#define WS_TOTAL_FLOATS 2793472   // 11,173,888 bytes
#define WS_OFF_L3       2752512   // float offset of level-3 plane (32768 floats)
#define WS_OFF_L4       2785280   // float offset of level-4 plane (8192 floats)
#define LDS_BYTES       163840    // levels 3+4 channel-last = 128KB + 32KB

// ---------------------------------------------------------------------------
// Pass 1: transpose (C,R,R) channel-major pyramid -> (R,R,C) channel-last ws.
// ---------------------------------------------------------------------------
__global__ __launch_bounds__(256) void nm_transpose(
    const float* __restrict__ t0, const float* __restrict__ t1,
    const float* __restrict__ t2, const float* __restrict__ t3,
    const float* __restrict__ t4, float* __restrict__ ws)
{
    int e = blockIdx.x * 256 + threadIdx.x;
    if (e >= WS_TOTAL_FLOATS) return;
    int l, base;
    if      (e < 2097152)   { l = 0; base = 0;         }
    else if (e < 2621440)   { l = 1; base = 2097152;   }
    else if (e < WS_OFF_L3) { l = 2; base = 2621440;   }
    else if (e < WS_OFF_L4) { l = 3; base = WS_OFF_L3; }
    else                    { l = 4; base = WS_OFF_L4; }
    const float* tex = (l == 0) ? t0 : (l == 1) ? t1 : (l == 2) ? t2
                      : (l == 3) ? t3 : t4;
    int r = e - base;
    int t = r >> 3;        // texel index (y*R + x)
    int c = r & 7;         // channel
    int R = 512 >> l;
    ws[e] = tex[c * R * R + t];
}

// ---------------------------------------------------------------------------
// Bilinear 8-channel tap from a channel-last plane (global ws OR LDS copy).
// Inlined at each call site so address-space inference specializes loads
// to global_load_b128 / ds_load_b128.
// ---------------------------------------------------------------------------
__device__ __forceinline__ void bilinear8(const float* base, int R,
                                          float cx, float cy, float w,
                                          float acc[CH])
{
    float x  = (cx + 1.0f) * 0.5f * (float)(R - 1);
    float y  = (cy + 1.0f) * 0.5f * (float)(R - 1);
    float xf = floorf(x), yf = floorf(y);
    float wx = x - xf,    wy = y - yf;
    int x0 = (int)xf, y0 = (int)yf;
    x0 = min(max(x0, 0), R - 2);
    y0 = min(max(y0, 0), R - 2);
    const float* p00 = base + (size_t)(y0 * R + x0) * CH;  // 32B-aligned texel
    const float* p10 = p00 + (size_t)R * CH;
    float w00 = (1.0f - wx) * (1.0f - wy);
    float w01 = wx * (1.0f - wy);
    float w10 = (1.0f - wx) * wy;
    float w11 = wx * wy;
    float4 a0 = ((const float4*)p00)[0];  // (x0,y0) ch0..3
    float4 a1 = ((const float4*)p00)[1];  // (x0,y0) ch4..7
    float4 b0 = ((const float4*)p00)[2];  // (x1,y0) ch0..3
    float4 b1 = ((const float4*)p00)[3];
    float4 c0 = ((const float4*)p10)[0];  // (x0,y1)
    float4 c1 = ((const float4*)p10)[1];
    float4 d0 = ((const float4*)p10)[2];  // (x1,y1)
    float4 d1 = ((const float4*)p10)[3];
    acc[0] += w * (w00 * a0.x + w01 * b0.x + w10 * c0.x + w11 * d0.x);
    acc[1] += w * (w00 * a0.y + w01 * b0.y + w10 * c0.y + w11 * d0.y);
    acc[2] += w * (w00 * a0.z + w01 * b0.z + w10 * c0.z + w11 * d0.z);
    acc[3] += w * (w00 * a0.w + w01 * b0.w + w10 * c0.w + w11 * d0.w);
    acc[4] += w * (w00 * a1.x + w01 * b1.x + w10 * c1.x + w11 * d1.x);
    acc[5] += w * (w00 * a1.y + w01 * b1.y + w10 * c1.y + w11 * d1.y);
    acc[6] += w * (w00 * a1.z + w01 * b1.z + w10 * c1.z + w11 * d1.z);
    acc[7] += w * (w00 * a1.w + w01 * b1.w + w10 * c1.w + w11 * d1.w);
}

// ---------------------------------------------------------------------------
// Pass 2: main sampling kernel.
//   LDS floats 0..32767 hold the level-3 plane (128KB)
//   LDS floats 32768..40959 hold the level-4 plane (32KB)
// Wave 0 first issues a TDM tensor_load_to_lds for the level-4 plane
// (TENSORcnt path); after s_wait_tensorcnt the per-lane async b128 loads
// stage the full 160KB (ASYNCcnt path), guaranteeing correct LDS contents
// independent of the uncharacterized 6th-operand semantics of the TDM
// builtin. Then grid-stride over pixels doing the trilinear blend.
// ---------------------------------------------------------------------------
__global__ __launch_bounds__(256) void nm_main(
    const float* __restrict__ uvs, const float* __restrict__ level,
    const float* __restrict__ ws, float* __restrict__ out)
{
    extern __shared__ float lds[];
    uint32_t lbase = (uint32_t)(uintptr_t)(&lds[0]);

#if __has_builtin(__builtin_amdgcn_tensor_load_to_lds)
    // TDM: DMA the 32KB level-4 plane into LDS (wave 0 issues; EXEC-independent)
    if (threadIdx.x < 32) {
        typedef unsigned int v4u __attribute__((ext_vector_type(4)));
        typedef int          v8i __attribute__((ext_vector_type(8)));
        typedef int          v4i __attribute__((ext_vector_type(4)));
        uint64_t ga = (uint64_t)(uintptr_t)(ws + WS_OFF_L4);
        uint32_t la = lbase + 131072;                // lds[32768] in bytes
        v4u g0;
        g0.x = 1u;                                   // count=1 valid descriptor
        g0.y = la;                                   // lds_addr (bytes)
        g0.z = (uint32_t)ga;                         // global_addr[31:0]
        g0.w = (uint32_t)((ga >> 32) & 0x01FFFFFFu)  // global_addr[56:32]
             | (2u << 30);                           // type=2 ("image")
        v8i g1;
        g1[0] = (int)(2u << 16);                     // data_size=2 -> 4 bytes
        g1[1] = (int)(8192u << 16);                  // tensor_dim0[15:0] << 16
        g1[2] = (int)(0u | (1u << 16));              // dim0 hi=0 | tensor_dim1=1
        g1[3] = (int)(0u | (8192u << 16));           // dim1 hi=0 | tile_dim0=8192
        g1[4] = 1;                                   // tile_dim1=1, tile_dim2=0
        g1[5] = 8192;                                // tensor_dim0_stride lo
        g1[6] = 0;                                   // stride0 hi | stride1 lo
        g1[7] = 0;                                   // stride1 hi
        v4i g2 = {0, 0, 0, 0};
        v4i g3 = {0, 0, 0, 0};
        v8i gx = {0, 0, 0, 0, 0, 0, 0, 0};           // 6-arg toolchain extra group
        __builtin_amdgcn_tensor_load_to_lds(g0, g1, g2, g3, gx, 0);
    }
    __builtin_amdgcn_s_wait_tensorcnt(0);
#endif

    // Async stage levels 3+4: 40960 floats = 10240 x 16B chunks, 40/thread.
    {
        const float* src = ws + WS_OFF_L3;   // levels 3,4 contiguous in ws
        for (int i = threadIdx.x; i < 10240; i += 256) {
            const float* g     = src + i * 4;
            uint32_t     laddr = lbase + i * 16;
            asm volatile("global_load_async_to_lds_b128 %0, %1, off"
                         :: "v"(laddr), "v"(g) : "memory");
        }
    }
    asm volatile("s_wait_asynccnt 0x0" ::: "memory");
    __syncthreads();

    const int stride = gridDim.x * 256;
    for (int p = blockIdx.x * 256 + threadIdx.x; p < NPIX; p += stride) {
        // streamed, read-once inputs -> nontemporal (keep pyramid L2-resident)
        float u  = __builtin_nontemporal_load(uvs + 2 * (size_t)p);
        float v  = __builtin_nontemporal_load(uvs + 2 * (size_t)p + 1);
        float lv = __builtin_nontemporal_load(level + p);

        float cx = (u - truncf(u)) * 2.0f - 1.0f;
        float cy = (v - truncf(v)) * 2.0f - 1.0f;

        float fb    = floorf(lv);
        float alpha = lv - fb;
        int lb = min(max((int)fb, 0), NLVL - 1);
        int ln = min(lb + 1, NLVL - 1);

        float acc[CH] = {0.f, 0.f, 0.f, 0.f, 0.f, 0.f, 0.f, 0.f};
        if (lb >= 3) bilinear8(lds + ((lb - 3) << 15), 512 >> lb, cx, cy, 1.0f - alpha, acc);
        else         bilinear8(ws + ws_off(lb),        512 >> lb, cx, cy, 1.0f - alpha, acc);
        if (ln >= 3) bilinear8(lds + ((ln - 3) << 15), 512 >> ln, cx, cy, alpha, acc);
        else         bilinear8(ws + ws_off(ln),        512 >> ln, cx, cy, alpha, acc);

        // out layout (B, CH, H, W); hw contiguous across lanes -> coalesced
        int    b  = p >> 20;            // p / HW
        size_t hw = (size_t)(p & (HW - 1));
        float* ob = out + (size_t)b * CH * HW + hw;
        #pragma unroll
        for (int c = 0; c < CH; ++c)
            __builtin_nontemporal_store(acc[c], ob + (size_t)c * HW);
    }
}

// ---------------------------------------------------------------------------
// Fallback: direct gathers from the original channel-major layout (used only
// if d_ws can't hold the transposed pyramid).
// ---------------------------------------------------------------------------
__device__ __forceinline__ void bilinear8_cm(const float* tex, int R,
                                             float cx, float cy, float w,
                                             float acc[CH])
{
    float x  = (cx + 1.0f) * 0.5f * (float)(R - 1);
    float y  = (cy + 1.0f) * 0.5f * (float)(R - 1);
    float xf = floorf(x), yf = floorf(y);
    float wx = x - xf,    wy = y - yf;
    int x0 = min(max((int)xf, 0), R - 2);
    int y0 = min(max((int)yf, 0), R - 2);
    float w00 = (1.0f - wx) * (1.0f - wy);
    float w01 = wx * (1.0f - wy);
    float w10 = (1.0f - wx) * wy;
    float w11 = wx * wy;
    int RR = R * R;
    const float* pp = tex + y0 * R + x0;
    #pragma unroll
    for (int c = 0; c < CH; ++c) {
        float v00 = pp[0], v01 = pp[1], v10 = pp[R], v11 = pp[R + 1];
        acc[c] += w * (w00 * v00 + w01 * v01 + w10 * v10 + w11 * v11);
        pp += RR;
    }
}

__global__ __launch_bounds__(256) void nm_direct(
    const float* __restrict__ uvs, const float* __restrict__ level,
    const float* __restrict__ t0, const float* __restrict__ t1,
    const float* __restrict__ t2, const float* __restrict__ t3,
    const float* __restrict__ t4, float* __restrict__ out)
{
    int p = blockIdx.x * 256 + threadIdx.x;
    if (p >= NPIX) return;
    float u  = uvs[2 * (size_t)p];
    float v  = uvs[2 * (size_t)p + 1];
    float lv = level[p];
    float cx = (u - truncf(u)) * 2.0f - 1.0f;
    float cy = (v - truncf(v)) * 2.0f - 1.0f;
    float fb    = floorf(lv);
    float alpha = lv - fb;
    int lb = min(max((int)fb, 0), NLVL - 1);
    int ln = min(lb + 1, NLVL - 1);

    float acc[CH] = {0.f, 0.f, 0.f, 0.f, 0.f, 0.f, 0.f, 0.f};
    const float* tb = (lb == 0) ? t0 : (lb == 1) ? t1 : (lb == 2) ? t2
                     : (lb == 3) ? t3 : t4;
    bilinear8_cm(tb, 512 >> lb, cx, cy, 1.0f - alpha, acc);
    const float* tn = (ln == 0) ? t0 : (ln == 1) ? t1 : (ln == 2) ? t2
                     : (ln == 3) ? t3 : t4;
    bilinear8_cm(tn, 512 >> ln, cx, cy, alpha, acc);

    int    b  = p >> 20;
    size_t hw = (size_t)(p & (HW - 1));
    float* ob = out + (size_t)b * CH * HW + hw;
    #pragma unroll
    for (int c = 0; c < CH; ++c)
        __builtin_nontemporal_store(acc[c], ob + (size_t)c * HW);
}

// ---------------------------------------------------------------------------
extern "C" void kernel_launch(void* const* d_in, const int* in_sizes, int n_in,
                              void* d_out, int out_size, void* d_ws, size_t ws_size,
                              hipStream_t stream)
{
    const float* uvs   = (const float*)d_in[0];
    const float* level = (const float*)d_in[1];
    const float* t0    = (const float*)d_in[2];
    const float* t1    = (const float*)d_in[3];
    const float* t2    = (const float*)d_in[4];
    const float* t3    = (const float*)d_in[5];
    const float* t4    = (const float*)d_in[6];
    float*       out   = (float*)d_out;

    const size_t need = (size_t)WS_TOTAL_FLOATS * sizeof(float);
    if (ws_size >= need) {
        float* ws = (float*)d_ws;
        (void)hipFuncSetAttribute((const void*)nm_main,
                                  hipFuncAttributeMaxDynamicSharedMemorySize,
                                  LDS_BYTES);
        nm_transpose<<<(WS_TOTAL_FLOATS + 255) / 256, 256, 0, stream>>>(
            t0, t1, t2, t3, t4, ws);
        nm_main<<<2048, 256, LDS_BYTES, stream>>>(uvs, level, ws, out);
    } else {
        nm_direct<<<(NPIX + 255) / 256, 256, 0, stream>>>(
            uvs, level, t0, t1, t2, t3, t4, out);
    }
}